// ScaledDotProductAttention_16269336117908
// MI455X (gfx1250) — compile-verified
//
#include <hip/hip_runtime.h>
#include <cstdint>
#include <cstddef>

typedef __attribute__((ext_vector_type(2))) float v2f;
typedef __attribute__((ext_vector_type(8))) float v8f;
typedef __attribute__((ext_vector_type(4))) int v4i;
typedef __attribute__((address_space(1))) v4i* gv4i_p;  // global AS
typedef __attribute__((address_space(3))) v4i* lv4i_p;  // LDS AS

namespace {
constexpr int kD      = 64;    // head dim
constexpr int kK      = 1024;  // key length
constexpr int kQ      = 1024;  // query length
constexpr int kQTile  = 16;    // queries per block
constexpr int kWaves  = 8;     // wave32 waves per block
constexpr int kThreads = kWaves * 32;
constexpr int kQPad   = kD + 4;   // LDS pad: rows land 4 banks apart
constexpr int kSPad   = kK + 4;   // 1028 % 64 == 4 -> conflict-free col reads
constexpr int kMaskBytes = kQTile * kK;  // 16 KB mask tile per block
constexpr float kNegInf = -1e9f;
}  // namespace

#if __has_builtin(__builtin_amdgcn_global_load_async_to_lds_b128)
#define SDPA_ASYNC_LDS 1
#else
#define SDPA_ASYNC_LDS 0
#endif

__global__ __launch_bounds__(kThreads) void sdpa_wmma_f32_kernel(
    const float* __restrict__ Qm, const float* __restrict__ Km,
    const float* __restrict__ Vm, const uint8_t* __restrict__ Mm,
    float* __restrict__ ctx_out, float* __restrict__ attn_out,
    int qtiles_per_b) {
  __shared__ float q_lds[kQTile][kQPad];            // scaled Q tile
  __shared__ float s_lds[kQTile][kSPad];            // scores -> probabilities
  __shared__ float c_lds[kQTile][kD];               // context partial (upper K half)
  __shared__ float red[kThreads];                   // softmax reduction scratch
  __shared__ float row_max[kQTile];
  __shared__ float row_sum[kQTile];
  __shared__ __align__(16) uint8_t m_lds[kQTile][kK];  // mask tile

  const int tid  = threadIdx.x;
  const int lane = tid & 31;
  const int wave = tid >> 5;
  const int lcol = lane & 15;   // A-row / B-col / C-col lane index
  const int lhi  = lane >> 4;   // selects K-pair (A/B) or row-half (C/D)

  const int b  = blockIdx.x / qtiles_per_b;
  const int q0 = (blockIdx.x % qtiles_per_b) * kQTile;

  const float* __restrict__ Qb = Qm + ((size_t)b * kQ + q0) * kD;
  const float* __restrict__ Kb = Km + (size_t)b * kK * kD;
  const float* __restrict__ Vb = Vm + (size_t)b * kK * kD;
  const uint8_t* __restrict__ Mb = Mm + ((size_t)b * kQ + q0) * kK;
  float* __restrict__ Ab = attn_out + ((size_t)b * kQ + q0) * kK;
  float* __restrict__ Cb = ctx_out  + ((size_t)b * kQ + q0) * kD;

  // ---- kick off async mask prefetch (HBM -> LDS) to overlap the S matmul ----
#if SDPA_ASYNC_LDS
  for (int off = tid * 16; off < kMaskBytes; off += kThreads * 16) {
    __builtin_amdgcn_global_load_async_to_lds_b128(
        (gv4i_p)(uintptr_t)(Mb + off),
        (lv4i_p)(&m_lds[0][0] + off),
        /*offset=*/0, /*cpol=*/0);
  }
#else
  for (int off = tid * 16; off < kMaskBytes; off += kThreads * 16) {
    *(uint4*)(&m_lds[0][0] + off) = *(const uint4*)(Mb + off);
  }
#endif

  // ---- stage Q tile, pre-scaled by 1/sqrt(64) ----
  for (int i = tid; i < kQTile * kD; i += kThreads) {
    q_lds[i / kD][i % kD] = Qb[i] * 0.125f;
  }
  __syncthreads();

  // ---- S = (Q/8) * K^T : each wave owns 8 of the 64 key tiles ----
  for (int kt = wave; kt < kK / 16; kt += kWaves) {
    const int key = kt * 16 + lcol;  // B-fragment column = key index
    v8f acc = {};
#pragma unroll
    for (int kc = 0; kc < kD / 4; ++kc) {
      const int kb = kc * 4 + lhi * 2;  // this lane's K-pair within the 4-chunk
      v2f a, bf;
      a.x  = q_lds[lcol][kb + 0];
      a.y  = q_lds[lcol][kb + 1];
      bf.x = Kb[(size_t)key * kD + kb + 0];  // K^T[kb][key] == K[key][kb]
      bf.y = Kb[(size_t)key * kD + kb + 1];
      acc = __builtin_amdgcn_wmma_f32_16x16x4_f32(
          false, a, false, bf, (short)0, acc, false, false);
    }
#pragma unroll
    for (int v = 0; v < 8; ++v) {
      s_lds[lhi * 8 + v][kt * 16 + lcol] = acc[v];
    }
  }

#if SDPA_ASYNC_LDS
  // each wave waits for its own async transfers; the barrier covers the rest
#if __has_builtin(__builtin_amdgcn_s_wait_asynccnt)
  __builtin_amdgcn_s_wait_asynccnt(0);
#else
  asm volatile("s_wait_asynccnt 0x0" ::: "memory");
#endif
#endif
  __syncthreads();

  // ---- masked softmax across each row (16 threads per row, 4-wide) ----
  {
    const int row = tid >> 4;
    const int sub = tid & 15;

    // pass 1: apply mask (LDS uchar4) + row max
    float m = -3.0e38f;
    for (int c = sub * 4; c < kK; c += 64) {
      float4 s4 = *(float4*)&s_lds[row][c];
      const uchar4 mk = *(const uchar4*)&m_lds[row][c];
      if (mk.x) s4.x = kNegInf;
      if (mk.y) s4.y = kNegInf;
      if (mk.z) s4.z = kNegInf;
      if (mk.w) s4.w = kNegInf;
      *(float4*)&s_lds[row][c] = s4;
      m = fmaxf(m, fmaxf(fmaxf(s4.x, s4.y), fmaxf(s4.z, s4.w)));
    }
    red[tid] = m;
    __syncthreads();
    if (tid < kQTile) {
      float mm = red[tid * 16];
      for (int j = 1; j < 16; ++j) mm = fmaxf(mm, red[tid * 16 + j]);
      row_max[tid] = mm;
    }
    __syncthreads();

    // pass 2: exponentiate + row sum
    const float rm = row_max[row];
    float s = 0.0f;
    for (int c = sub * 4; c < kK; c += 64) {
      float4 v = *(float4*)&s_lds[row][c];
      v.x = __expf(v.x - rm);
      v.y = __expf(v.y - rm);
      v.z = __expf(v.z - rm);
      v.w = __expf(v.w - rm);
      *(float4*)&s_lds[row][c] = v;
      s += (v.x + v.y) + (v.z + v.w);
    }
    red[tid] = s;
    __syncthreads();
    if (tid < kQTile) {
      float ss = 0.0f;
      for (int j = 0; j < 16; ++j) ss += red[tid * 16 + j];
      row_sum[tid] = ss;
    }
    __syncthreads();

    // pass 3: normalize, keep P in LDS, stream attn to HBM as b128 stores
    const float inv = 1.0f / row_sum[row];
    for (int c = sub * 4; c < kK; c += 64) {
      float4 v = *(float4*)&s_lds[row][c];
      v.x *= inv; v.y *= inv; v.z *= inv; v.w *= inv;
      *(float4*)&s_lds[row][c] = v;
      *(float4*)&Ab[(size_t)row * kK + c] = v;
    }
  }
  __syncthreads();

  // ---- context = P (16x1024) * V (1024x64) ----
  // waves 0-3: column tiles 0-3 over K in [0,512)
  // waves 4-7: column tiles 0-3 over K in [512,1024); reduced via LDS
  {
    const int ct = wave & 3;
    const int kh = wave >> 2;
    const int dcol = ct * 16 + lcol;
    v8f acc = {};
    for (int i = 0; i < (kK / 2) / 4; ++i) {
      const int kk = kh * (kK / 2) + i * 4 + lhi * 2;
      v2f a, bf;
      a.x  = s_lds[lcol][kk + 0];
      a.y  = s_lds[lcol][kk + 1];
      bf.x = Vb[(size_t)(kk + 0) * kD + dcol];
      bf.y = Vb[(size_t)(kk + 1) * kD + dcol];
      acc = __builtin_amdgcn_wmma_f32_16x16x4_f32(
          false, a, false, bf, (short)0, acc, false, false);
    }
    if (kh == 1) {
#pragma unroll
      for (int v = 0; v < 8; ++v) c_lds[lhi * 8 + v][dcol] = acc[v];
    }
    __syncthreads();
    if (kh == 0) {
#pragma unroll
      for (int v = 0; v < 8; ++v) {
        const int row = lhi * 8 + v;
        Cb[(size_t)row * kD + dcol] = acc[v] + c_lds[row][dcol];
      }
    }
  }
}

extern "C" void kernel_launch(void* const* d_in, const int* in_sizes, int n_in,
                              void* d_out, int out_size, void* d_ws, size_t ws_size,
                              hipStream_t stream) {
  (void)n_in; (void)out_size; (void)d_ws; (void)ws_size;

  const float*   Qm = (const float*)d_in[0];
  const float*   Km = (const float*)d_in[1];
  const float*   Vm = (const float*)d_in[2];
  const uint8_t* Mm = (const uint8_t*)d_in[3];

  const int B = in_sizes[0] / (kQ * kD);  // 32
  float* ctx_out  = (float*)d_out;                      // [B, Q, D]
  float* attn_out = ctx_out + (size_t)B * kQ * kD;      // [B, Q, K]

  const int qtiles_per_b = kQ / kQTile;                 // 64
  dim3 grid(B * qtiles_per_b);                          // 2048 blocks
  dim3 block(kThreads);                                 // 8 wave32 waves
  sdpa_wmma_f32_kernel<<<grid, block, 0, stream>>>(
      Qm, Km, Vm, Mm, ctx_out, attn_out, qtiles_per_b);
}